// Correlation_28810640621986
// MI455X (gfx1250) — compile-verified
//
#include <hip/hip_runtime.h>
#include <math.h>

// CDNA5 / gfx1250, wave32.
// Correlation kernel: per-window (10 x 128) centered Gram matrix via
// V_WMMA_F32_16X16X4_F32, strictly-lower-triangle output with divide_no_nan.
//
// Key identity: with std = sqrt(var/10) and cov = c/10 (c = raw Gram sum),
//   cov / (std_i * std_j) = c * rsqrt(var_i) * rsqrt(var_j)
// so no division or sqrt is needed per output element; divide_no_nan is
// implemented exactly by rstd[f] = (var > 0) ? rsqrt(var) : 0.

typedef __attribute__((ext_vector_type(2))) float v2f;
typedef __attribute__((ext_vector_type(8))) float v8f;

#define WSTRIDE 10     // window length (samples)
#define NFEAT   128    // features
#define KPAD    12     // pad K 10 -> 12 (3 WMMA steps of K=4; pad rows are zero)
#define NTILES_LOWER 36  // 8x8 tile grid, lower triangle incl. diagonal
#define OUT_PER_WIN (NFEAT * (NFEAT - 1) / 2)  // 8128

__global__ __launch_bounds__(128) void
Correlation_28810640621986_kernel(const float* __restrict__ x,
                                  float* __restrict__ out,
                                  int n_win) {
  __shared__ float ms[KPAD][NFEAT];   // centered window, K-major (12 x 128), 6 KB
  __shared__ float rstd[NFEAT];       // rsqrt(var) per feature (0 if var==0)

  const int win = blockIdx.x;
  if (win >= n_win) return;
  const int tid = threadIdx.x;        // 0..127, one feature per thread

  // ---- Phase 1: center window, compute rsqrt(var) -------------------------
  // Window (b,l) is contiguous: offset = win * (WSTRIDE*NFEAT).
  const float* xw = x + (size_t)win * (WSTRIDE * NFEAT);
  float v[WSTRIDE];
  float sum = 0.0f;
#pragma unroll
  for (int s = 0; s < WSTRIDE; ++s) {
    v[s] = xw[s * NFEAT + tid];       // coalesced: lanes hit consecutive floats
    sum += v[s];
  }
  const float mean = sum * (1.0f / WSTRIDE);
  float var = 0.0f;                   // raw sum of squares (scale folded below)
#pragma unroll
  for (int s = 0; s < WSTRIDE; ++s) {
    const float d = v[s] - mean;
    ms[s][tid] = d;
    var += d * d;
  }
  ms[10][tid] = 0.0f;                 // zero K-padding rows: contribute nothing
  ms[11][tid] = 0.0f;
  rstd[tid] = (var > 0.0f) ? rsqrtf(var) : 0.0f;  // v_rsq_f32; 0 => divide_no_nan
  __syncthreads();

  // ---- Phase 2: lower-triangle Gram tiles via WMMA ------------------------
  const int lane    = tid & 31;
  const int wave    = tid >> 5;       // 0..3
  const int halfsel = lane >> 4;      // 0: lanes 0-15, 1: lanes 16-31
  const int l15     = lane & 15;

  float* outw = out + (size_t)win * OUT_PER_WIN;

  for (int t = wave; t < NTILES_LOWER; t += 4) {
    // decode t -> (bi, bj), row-major over lower triangle incl. diagonal
    int bi = 0;
    while (((bi + 1) * (bi + 2)) / 2 <= t) ++bi;   // wave-uniform scalar loop
    const int bj = t - (bi * (bi + 1)) / 2;

    // D(16x16) = sum_k A(16x4) x B(4x16); A = ms^T rows [bi*16..), B = ms cols [bj*16..)
    // 32-bit A 16x4 layout: lanes 0-15 -> M=lane, VGPR{0,1}=K{0,1};
    //                       lanes 16-31 -> M=lane-16, VGPR{0,1}=K{2,3}.
    v8f c = {};
#pragma unroll
    for (int kb = 0; kb < KPAD; kb += 4) {
      const int k0 = kb + halfsel * 2;
      v2f a, b;
      a.x = ms[k0 + 0][bi * 16 + l15];
      a.y = ms[k0 + 1][bi * 16 + l15];
      b.x = ms[k0 + 0][bj * 16 + l15];
      b.y = ms[k0 + 1][bj * 16 + l15];
      // 8 args: (neg_a, A, neg_b, B, c_mod, C, reuse_a, reuse_b)
      c = __builtin_amdgcn_wmma_f32_16x16x4_f32(
          false, a, false, b, (short)0, c, false, false);
    }

    // C/D layout: VGPR r, lanes 0-15 -> M=r, N=lane; lanes 16-31 -> M=r+8.
    const int jg   = bj * 16 + l15;
    const float rj = rstd[jg];

    if (bi != bj) {
      // Strictly below-diagonal tile: every element is in the lower triangle.
#pragma unroll
      for (int r = 0; r < 8; ++r) {
        const int ig = bi * 16 + r + halfsel * 8;
        const float res = c[r] * rstd[ig] * rj;
        __builtin_nontemporal_store(res, &outw[(size_t)(ig * (ig - 1) / 2) + jg]);
      }
    } else {
      // Diagonal tile: only elements with ig > jg.
#pragma unroll
      for (int r = 0; r < 8; ++r) {
        const int ig = bi * 16 + r + halfsel * 8;
        if (ig > jg) {
          const float res = c[r] * rstd[ig] * rj;
          __builtin_nontemporal_store(res, &outw[(size_t)(ig * (ig - 1) / 2) + jg]);
        }
      }
    }
  }
}

extern "C" void kernel_launch(void* const* d_in, const int* in_sizes, int n_in,
                              void* d_out, int out_size, void* d_ws, size_t ws_size,
                              hipStream_t stream) {
  (void)n_in; (void)out_size; (void)d_ws; (void)ws_size;
  const float* x = (const float*)d_in[0];
  float* out = (float*)d_out;
  const int n_win = in_sizes[0] / (WSTRIDE * NFEAT);  // 64*1000*128 / 1280 = 6400
  Correlation_28810640621986_kernel<<<n_win, 128, 0, stream>>>(x, out, n_win);
}